// GaussianRenderer_43748536877199
// MI455X (gfx1250) — compile-verified
//
#include <hip/hip_runtime.h>
#include <cstdint>
#include <cstddef>

// ---------------- problem constants (match reference) ----------------
#define NG      8192
#define IMG_H   256
#define IMG_W   256
#define FX      300.0f
#define FY      300.0f
#define CX      128.0f
#define CY      128.0f
#define TTHRESH 1e-4f

#define REC_F    12                 // floats per packed gaussian record (48 B)
#define CHUNK_G  256                // gaussians staged per LDS chunk
#define CHUNK_F4 (CHUNK_G * 3)      // float4 units per chunk (768)
#define NCHUNK   (NG / CHUNK_G)     // 32

// exp(-0.5*q) == exp2(q * -log2(e)/2)
#define NHALF_LOG2E 0.7213475204444817f

typedef __attribute__((ext_vector_type(2))) float v2f;
typedef __attribute__((ext_vector_type(8))) float v8f;

// =====================================================================
// Kernel 1: per-gaussian preprocess.
// Emits, per gaussian, the 6-feature row of the exponent matmul
//   e(px,py) = a0*px^2 + a1*px*py + a2*py^2 + a3*px + a4*py + a5
// (conic pre-scaled by -log2(e)/2 so e feeds v_exp_f32 directly),
// plus color and sigmoid(alpha), and the stable depth sort key.
// Record: [a0 a1 a2 a3 a4 a5 0 0 | r g b alpha]  (48 B)
// =====================================================================
__global__ void gs_preprocess(const float* __restrict__ mean,
                              const float* __restrict__ qvec,
                              const float* __restrict__ lsv,
                              const float* __restrict__ color,
                              const float* __restrict__ alpha,
                              const float* __restrict__ c2w,
                              unsigned long long* __restrict__ keys,
                              float* __restrict__ rec)
{
    int g = blockIdx.x * blockDim.x + threadIdx.x;
    if (g >= NG) return;

    // Wm = R^T, t = -c2w[:,3]   (c2w is 3x4 row-major)
    float Wm[3][3], t[3];
    #pragma unroll
    for (int i = 0; i < 3; ++i) {
        #pragma unroll
        for (int j = 0; j < 3; ++j) Wm[i][j] = c2w[j * 4 + i];
        t[i] = -c2w[i * 4 + 3];
    }

    float pm[3], p[3];
    #pragma unroll
    for (int j = 0; j < 3; ++j) pm[j] = mean[g * 3 + j] + t[j];
    #pragma unroll
    for (int i = 0; i < 3; ++i)
        p[i] = Wm[i][0] * pm[0] + Wm[i][1] * pm[1] + Wm[i][2] * pm[2];

    float sv[3];
    #pragma unroll
    for (int j = 0; j < 3; ++j) sv[j] = expf(lsv[g * 3 + j]);

    float qw = qvec[g * 4 + 0], qx = qvec[g * 4 + 1];
    float qy = qvec[g * 4 + 2], qz = qvec[g * 4 + 3];
    float qinv = 1.0f / sqrtf(qw * qw + qx * qx + qy * qy + qz * qz);
    qw *= qinv; qx *= qinv; qy *= qinv; qz *= qinv;

    float R[3][3] = {
        {1.0f - 2.0f * (qy * qy + qz * qz), 2.0f * (qx * qy - qw * qz), 2.0f * (qx * qz + qw * qy)},
        {2.0f * (qx * qy + qw * qz), 1.0f - 2.0f * (qx * qx + qz * qz), 2.0f * (qy * qz - qw * qx)},
        {2.0f * (qx * qz - qw * qy), 2.0f * (qy * qz + qw * qx), 1.0f - 2.0f * (qx * qx + qy * qy)}};

    float M[3][3], S[3][3];
    #pragma unroll
    for (int i = 0; i < 3; ++i)
        #pragma unroll
        for (int j = 0; j < 3; ++j) M[i][j] = R[i][j] * sv[j];
    #pragma unroll
    for (int i = 0; i < 3; ++i)
        #pragma unroll
        for (int j = 0; j < 3; ++j)
            S[i][j] = M[i][0] * M[j][0] + M[i][1] * M[j][1] + M[i][2] * M[j][2];

    float x = p[0], y = p[1], z = p[2];
    float iz = 1.0f / z, iz2 = iz * iz;
    float J0[3] = {iz, 0.0f, -x * iz2};
    float J1[3] = {0.0f, iz, -y * iz2};

    float JW0[3], JW1[3];
    #pragma unroll
    for (int j = 0; j < 3; ++j) {
        JW0[j] = J0[0] * Wm[0][j] + J0[1] * Wm[1][j] + J0[2] * Wm[2][j];
        JW1[j] = J1[0] * Wm[0][j] + J1[1] * Wm[1][j] + J1[2] * Wm[2][j];
    }

    float T0[3], T1[3];
    #pragma unroll
    for (int j = 0; j < 3; ++j) {
        T0[j] = S[j][0] * JW0[0] + S[j][1] * JW0[1] + S[j][2] * JW0[2];
        T1[j] = S[j][0] * JW1[0] + S[j][1] * JW1[1] + S[j][2] * JW1[2];
    }
    float a = JW0[0] * T0[0] + JW0[1] * T0[1] + JW0[2] * T0[2];
    float b = JW1[0] * T0[0] + JW1[1] * T0[1] + JW1[2] * T0[2];
    float c = JW1[0] * T1[0] + JW1[1] * T1[1] + JW1[2] * T1[2];

    float det  = fmaxf(a * c - b * b, 1e-12f);
    float idet = 1.0f / det;

    float alph = 1.0f / (1.0f + expf(-alpha[g]));

    // stable ascending-depth key: depth > 0 so float bits are monotonic
    keys[g] = ((unsigned long long)__float_as_uint(z) << 32) | (unsigned)g;

    float mx = x * iz, my = y * iz;
    float k0 = (c * idet)  * (-NHALF_LOG2E);
    float k1 = (-b * idet) * (-2.0f * NHALF_LOG2E);
    float k2 = (a * idet)  * (-NHALF_LOG2E);

    float* r = rec + (size_t)g * REC_F;
    r[0]  = k0;                                    // * px^2
    r[1]  = k1;                                    // * px*py
    r[2]  = k2;                                    // * py^2
    r[3]  = -(2.0f * k0 * mx + k1 * my);           // * px
    r[4]  = -(k1 * mx + 2.0f * k2 * my);           // * py
    r[5]  = k0 * mx * mx + k1 * mx * my + k2 * my * my;  // * 1
    r[6]  = 0.0f; r[7] = 0.0f;                     // K-pad (features 6,7)
    r[8]  = color[g * 3 + 0];
    r[9]  = color[g * 3 + 1];
    r[10] = color[g * 3 + 2];
    r[11] = alph;
}

// =====================================================================
// Kernel 2: full bitonic sort of 8192 u64 keys in one workgroup's LDS.
// =====================================================================
__global__ void __launch_bounds__(1024) gs_sort(unsigned long long* __restrict__ keys)
{
    __shared__ unsigned long long sk[NG];    // 64 KB (WGP has 320 KB)

    for (int i = threadIdx.x; i < NG; i += 1024) sk[i] = keys[i];
    __syncthreads();

    for (int k = 2; k <= NG; k <<= 1) {
        for (int j = k >> 1; j > 0; j >>= 1) {
            #pragma unroll 1
            for (int t = threadIdx.x; t < NG / 2; t += 1024) {
                int i = ((t & ~(j - 1)) << 1) | (t & (j - 1));  // bit j clear
                unsigned long long va = sk[i];
                unsigned long long vb = sk[i + j];
                bool up = ((i & k) == 0);
                if ((va > vb) == up) { sk[i] = vb; sk[i + j] = va; }
            }
            __syncthreads();
        }
    }

    for (int i = threadIdx.x; i < NG; i += 1024) keys[i] = sk[i];
}

// =====================================================================
// Kernel 3: gather records into sorted (ascending depth) order
// =====================================================================
__global__ void gs_gather(const unsigned long long* __restrict__ keys,
                          const float4* __restrict__ rec,
                          float4* __restrict__ srec)
{
    int i = blockIdx.x * blockDim.x + threadIdx.x;
    if (i >= NG) return;
    int idx = (int)(keys[i] & 0xffffffffULL);
    #pragma unroll
    for (int k = 0; k < 3; ++k) srec[i * 3 + k] = rec[idx * 3 + k];
}

// =====================================================================
// Kernel 4: compositing. One thread per pixel; chunks async-staged into
// LDS (ASYNCcnt); exponents of 16 gaussians x 32 pixels per wave are
// computed with V_WMMA_F32_16X16X4_F32 (K=6 features padded to 8).
// =====================================================================
__global__ void __launch_bounds__(256) gs_render(const float* __restrict__ srec,
                                                 float* __restrict__ out)
{
    __shared__ float4 sbuf[2][CHUNK_F4];          // 2 x 12 KB

    const int p    = blockIdx.x * 256 + threadIdx.x;  // 0..65535
    const int wpx  = p & (IMG_W - 1);
    const int hpx  = p >> 8;
    const int lane = threadIdx.x & 31;
    const bool laneHi = (lane & 16) != 0;

    const float px = (-CX / FX) + ((float)wpx + 0.5f) * (1.0f / FX);
    const float py = (-CY / FY) + ((float)hpx + 0.5f) * (1.0f / FY);

    // ---- per-pixel feature vector b = [px^2, px*py, py^2, px, py, 1, 0, 0]
    const float bx2 = px * px, bxy = px * py, by2 = py * py;
    // partner-half features (built once)
    const float sx2 = __shfl_xor(bx2, 16, 32);
    const float sxy = __shfl_xor(bxy, 16, 32);
    const float sy2 = __shfl_xor(by2, 16, 32);
    const float spx = __shfl_xor(px, 16, 32);
    const float spy = __shfl_xor(py, 16, 32);

    // B operands (4x16 f32: V0 = K0(lanes0-15)/K2(lanes16-31), V1 = K1/K3)
    // tile0 = pixels of lanes 0-15, tile1 = pixels of lanes 16-31
    v2f B1t0, B2t0, B1t1, B2t1;
    B1t0.x = laneHi ? sy2 : bx2;  B1t0.y = laneHi ? spx : bxy;   // K0..K3
    B2t0.x = laneHi ? 0.0f : py;  B2t0.y = laneHi ? 0.0f : 1.0f; // K4..K7
    B1t1.x = laneHi ? by2 : sx2;  B1t1.y = laneHi ? px : sxy;
    B2t1.x = laneHi ? 0.0f : spy; B2t1.y = laneHi ? 0.0f : 1.0f;

    float T = 1.0f, r0 = 0.0f, r1 = 0.0f, r2 = 0.0f;

    // async-stage one 256-gaussian chunk (768 x b128) into LDS buffer `buf`
    auto issue_chunk = [&](int chunk, int buf) {
        const char* gbase = (const char*)srec + (size_t)chunk * (CHUNK_G * REC_F * 4);
        #pragma unroll
        for (int it = 0; it < 3; ++it) {
            unsigned u    = (unsigned)threadIdx.x + 256u * it;      // 0..767
            unsigned lds  = (unsigned)(uintptr_t)(&sbuf[buf][u]);   // LDS byte addr
            unsigned voff = u * 16u;                                 // global byte off
            asm volatile("global_load_async_to_lds_b128 %0, %1, %2"
                         :: "v"(lds), "v"(voff), "s"(gbase) : "memory");
        }
    };

    issue_chunk(0, 0);

    for (int c = 0; c < NCHUNK; ++c) {
        const int buf = c & 1;
        if (c + 1 < NCHUNK) {
            issue_chunk(c + 1, buf ^ 1);
            // chunk c's 3 loads are the oldest of <=6 outstanding; async loads
            // complete in order, so draining to <=3 completes exactly chunk c.
            asm volatile("s_wait_asynccnt 0x3" ::: "memory");
        } else {
            asm volatile("s_wait_asynccnt 0x0" ::: "memory");
        }
        __syncthreads();

        // wave-uniform skip keeps EXEC all-1s inside (WMMA requirement);
        // T is monotonically non-increasing so skipping is safe.
        if (__any(T > TTHRESH)) {
            const float* sA = (const float*)&sbuf[buf][0];

            #pragma unroll 1
            for (int grp = 0; grp < CHUNK_G / 16; ++grp) {
                // ---- A operands: 16 gaussians x features, straight from LDS.
                // A 16x4 f32: lanes0-15 = rows, V0=K0,V1=K1; lanes16-31: K2,K3.
                const float* ga = sA + (grp * 16 + (lane & 15)) * REC_F
                                     + (laneHi ? 2 : 0);
                v2f A1; A1.x = ga[0]; A1.y = ga[1];   // feats 0,1 / 2,3
                v2f A2; A2.x = ga[4]; A2.y = ga[5];   // feats 4,5 / 6,7(=0 pad)

                // ---- exponents: E[16 gaussians x 16 pixels] per tile
                v8f acc0 = {};
                acc0 = __builtin_amdgcn_wmma_f32_16x16x4_f32(
                           false, A1, false, B1t0, (short)0, acc0, false, false);
                acc0 = __builtin_amdgcn_wmma_f32_16x16x4_f32(
                           false, A2, false, B2t0, (short)0, acc0, false, false);
                v8f acc1 = {};
                acc1 = __builtin_amdgcn_wmma_f32_16x16x4_f32(
                           false, A1, false, B1t1, (short)0, acc1, false, false);
                acc1 = __builtin_amdgcn_wmma_f32_16x16x4_f32(
                           false, A2, false, B2t1, (short)0, acc1, false, false);

                // ---- half-wave swap so each lane owns its pixel's 16 exps.
                // lane<16 owns D rows 0-7 of tile0 (its pixel), needs rows 8-15
                // (held by lane+16); symmetric for tile1 / lanes 16-31.
                float sexp[8];
                #pragma unroll
                for (int r = 0; r < 8; ++r) {
                    float u = laneHi ? acc0[r] : acc1[r];   // value partner needs
                    sexp[r] = __shfl_xor(u, 16, 32);
                }

                // ---- sequential front-to-back blend over the 16 gaussians
                #pragma unroll
                for (int i = 0; i < 16; ++i) {
                    float ei = (i < 8) ? (laneHi ? sexp[i] : acc0[i])
                                       : (laneHi ? acc1[i - 8] : sexp[i - 8]);
                    const float* gc = sA + (grp * 16 + i) * REC_F + 8; // r,g,b,alpha
                    // raw v_exp_f32: exponent <= 0 here; inputs < -126 flush to
                    // 0, which is exactly a zero-weight contribution anyway.
                    float wgt = fminf(gc[3] * __builtin_amdgcn_exp2f(ei), 0.9999f);
                    float ct  = (T > TTHRESH) ? T * wgt : 0.0f;
                    r0 = fmaf(ct, gc[0], r0);
                    r1 = fmaf(ct, gc[1], r1);
                    r2 = fmaf(ct, gc[2], r2);
                    T *= (1.0f - wgt);
                }
            }
        }
        // barrier doubles as "safe to overwrite buf" fence + block-wide vote
        if (__syncthreads_and(T <= TTHRESH)) break;
    }

    out[p * 3 + 0] = r0;
    out[p * 3 + 1] = r1;
    out[p * 3 + 2] = r2;
}

// =====================================================================
// host-side launcher (4 graph nodes total)
// =====================================================================
extern "C" void kernel_launch(void* const* d_in, const int* in_sizes, int n_in,
                              void* d_out, int out_size, void* d_ws, size_t ws_size,
                              hipStream_t stream)
{
    (void)in_sizes; (void)n_in; (void)out_size; (void)ws_size;

    const float* mean  = (const float*)d_in[0];   // (N,3)
    const float* qvec  = (const float*)d_in[1];   // (N,4)
    const float* lsv   = (const float*)d_in[2];   // (N,3)
    const float* color = (const float*)d_in[3];   // (N,3)
    const float* alpha = (const float*)d_in[4];   // (N,)
    const float* c2w   = (const float*)d_in[5];   // (3,4)

    float* out = (float*)d_out;                   // (H,W,3) f32

    unsigned long long* keys = (unsigned long long*)d_ws;                          // 64 KB
    float* rec  = (float*)((char*)d_ws + (size_t)NG * 8);                          // 384 KB
    float* srec = (float*)((char*)d_ws + (size_t)NG * 8 + (size_t)NG * REC_F * 4); // 384 KB

    const int TB = 256;
    const int GB = NG / TB;   // 32

    gs_preprocess<<<GB, TB, 0, stream>>>(mean, qvec, lsv, color, alpha, c2w, keys, rec);
    gs_sort<<<1, 1024, 0, stream>>>(keys);
    gs_gather<<<GB, TB, 0, stream>>>(keys, (const float4*)rec, (float4*)srec);
    gs_render<<<(IMG_H * IMG_W) / TB, TB, 0, stream>>>(srec, out);
}